// CorefDecoderWL_3444563771556
// MI455X (gfx1250) — compile-verified
//
#include <hip/hip_runtime.h>

#define HIDDEN 1024

typedef __attribute__((ext_vector_type(2))) float v2f;
typedef __attribute__((ext_vector_type(8))) float v8f;

// ---------------------------------------------------------------------------
// Kernel 1: scores[i] = dot(hidden[i], w) + b   (one wave32 per subword row)
// ---------------------------------------------------------------------------
__global__ void coref_scores_kernel(const float* __restrict__ hs,
                                    const float* __restrict__ w,
                                    const float* __restrict__ b,
                                    float* __restrict__ scores,
                                    int n_sub) {
    const int lane = threadIdx.x & 31;
    const int wave = threadIdx.x >> 5;
    const int row  = blockIdx.x * 8 + wave;
    if (row >= n_sub) return;

    const float4* h4 = (const float4*)(hs + (size_t)row * HIDDEN);
    const float4* w4 = (const float4*)w;

    float s = 0.f;
#pragma unroll
    for (int j = 0; j < HIDDEN / 128; ++j) {      // 8 iters: 32 lanes * float4 = 128 floats/iter
        float4 a = h4[lane + 32 * j];
        float4 c = w4[lane + 32 * j];
        s += a.x * c.x + a.y * c.y + a.z * c.z + a.w * c.w;
    }
#pragma unroll
    for (int off = 16; off > 0; off >>= 1) s += __shfl_xor(s, off, 32);

    if (lane == 0) scores[row] = s + b[0];
}

// ---------------------------------------------------------------------------
// Kernel 2: per 16-word group, out[16 x 1024] = A[16 x K] * hidden[kbeg..kend]
// A holds per-word softmax weights over the union span; computed via WMMA
// V_WMMA_F32_16X16X4_F32 stepping K by 4.  Block = 256 threads = 8 waves;
// wave j owns hidden columns [j*128, j*128+128) as eight 16-wide tiles.
// ---------------------------------------------------------------------------
__global__ void coref_pool_wmma_kernel(const float* __restrict__ hs,
                                       const long long* __restrict__ starts,
                                       const long long* __restrict__ ends,
                                       const float* __restrict__ scores,
                                       float* __restrict__ out,
                                       int n_words, int n_sub) {
    __shared__ int   s_start[16];
    __shared__ int   s_end[16];
    __shared__ float s_max[16];
    __shared__ float s_inv[16];
    __shared__ int   s_kb, s_ke;

    const int tid = threadIdx.x;
    const int w0  = blockIdx.x * 16;

    // Per-word softmax statistics (spans are short; serial per-word loops).
    if (tid < 16) {
        const int wi = w0 + tid;
        int st = 0, en = -1;                       // invalid word -> empty span
        if (wi < n_words) {
            st = (int)starts[wi];
            en = (int)ends[wi];
        }
        s_start[tid] = st;
        s_end[tid]   = en;
        float m = -3.402823466e+38f;
        for (int j = st; j <= en; ++j) m = fmaxf(m, scores[j]);
        float sum = 0.f;
        for (int j = st; j <= en; ++j) sum += __expf(scores[j] - m);
        s_max[tid] = m;
        s_inv[tid] = (sum > 0.f) ? (1.0f / sum) : 0.f;
    }
    __syncthreads();
    if (tid == 0) {
        int kb = 0x7fffffff, ke = -1;
        for (int i = 0; i < 16; ++i) {
            if (s_end[i] >= s_start[i]) {
                kb = (s_start[i] < kb) ? s_start[i] : kb;
                ke = (s_end[i]   > ke) ? s_end[i]   : ke;
            }
        }
        if (ke < 0) kb = 0;
        s_kb = kb;
        s_ke = ke;
    }
    __syncthreads();

    const int kbeg = s_kb;
    const int K    = s_ke - s_kb + 1;              // 0 if no valid word in group

    const int lane  = tid & 31;
    const int wave  = tid >> 5;
    const int m     = lane & 15;                   // A-row (word) / B,D column-in-tile
    const int half  = lane >> 4;                   // 0 or 1

    const int   st  = s_start[m];
    const int   en  = s_end[m];
    const float mx  = s_max[m];
    const float inv = s_inv[m];

    const int ncol0 = wave * 128;                  // this wave's column range

    v8f acc[8] = {};

    for (int k = 0; k < K; k += 4) {
        // A 16x4 f32 layout: VGPR0 -> K = k + half*2, VGPR1 -> K = k + half*2 + 1
        const int kk0 = kbeg + k + half * 2;
        const int kk1 = kk0 + 1;
        v2f a;
        a.x = (kk0 >= st && kk0 <= en) ? __expf(scores[kk0] - mx) * inv : 0.f;
        a.y = (kk1 >= st && kk1 <= en) ? __expf(scores[kk1] - mx) * inv : 0.f;

        // B 4x16 layout: VGPR0 -> row K = k + half*2, VGPR1 -> row K = k + half*2 + 1
        int rb0 = kk0; if (rb0 > n_sub - 1) rb0 = n_sub - 1;   // padded K: weight is 0
        int rb1 = kk1; if (rb1 > n_sub - 1) rb1 = n_sub - 1;
        const float* B0 = hs + (size_t)rb0 * HIDDEN;
        const float* B1 = hs + (size_t)rb1 * HIDDEN;

#pragma unroll
        for (int t = 0; t < 8; ++t) {
            const int col = ncol0 + t * 16 + m;
            v2f bb;
            bb.x = B0[col];
            bb.y = B1[col];
            acc[t] = __builtin_amdgcn_wmma_f32_16x16x4_f32(
                false, a, false, bb, (short)0, acc[t], false, false);
        }
    }

    // D 16x16 f32 layout: VGPR v -> M = v + half*8, N = lane&15
#pragma unroll
    for (int t = 0; t < 8; ++t) {
        const int col = ncol0 + t * 16 + m;
#pragma unroll
        for (int v = 0; v < 8; ++v) {
            const int row = w0 + v + half * 8;
            if (row < n_words) out[(size_t)row * HIDDEN + col] = acc[t][v];
        }
    }
}

// ---------------------------------------------------------------------------
extern "C" void kernel_launch(void* const* d_in, const int* in_sizes, int n_in,
                              void* d_out, int out_size, void* d_ws, size_t ws_size,
                              hipStream_t stream) {
    const float*     hs     = (const float*)d_in[0];
    const long long* starts = (const long long*)d_in[1];
    const long long* ends   = (const long long*)d_in[2];
    const float*     w      = (const float*)d_in[3];
    const float*     b      = (const float*)d_in[4];
    float*           out    = (float*)d_out;
    float*           scores = (float*)d_ws;       // n_sub floats of scratch

    const int n_sub   = in_sizes[0] / HIDDEN;
    const int n_words = in_sizes[1];

    coref_scores_kernel<<<(n_sub + 7) / 8, 256, 0, stream>>>(hs, w, b, scores, n_sub);

    const int n_groups = (n_words + 15) / 16;
    coref_pool_wmma_kernel<<<n_groups, 256, 0, stream>>>(hs, starts, ends, scores,
                                                         out, n_words, n_sub);
}